// FieldModel_13795434955366
// MI455X (gfx1250) — compile-verified
//
#include <hip/hip_runtime.h>
#include <math.h>

#define BATCH   4
#define N_ATOMS 4096
#define MAX_NEI 100
#define N_GHOST 1024
#define N_ALL   (N_ATOMS + N_GHOST)
#define D_EMB   32
#define D_HID   64
#define CUTOFF  6.0f
#define PI_F    3.14159265358979323846f

typedef float v2f __attribute__((ext_vector_type(2)));
typedef float v8f __attribute__((ext_vector_type(8)));

// d_out layout (flat float32, reference return order):
// [Etot(4) | Ei(16384) | Force(4*5120*3) | Virial(36) | virial(4*4096*9)]
#define OFF_ETOT        0
#define OFF_EI          4
#define OFF_FORCE       (OFF_EI + BATCH * N_ATOMS)
#define OFF_VIRIAL      (OFF_FORCE + BATCH * N_ALL * 3)
#define OFF_VIRIAL_PER  (OFF_VIRIAL + BATCH * 9)
#define N_ZERO_TAIL     (BATCH * N_ALL * 3 + BATCH * 9)   // Force+Virial contiguous

// ---------------------------------------------------------------- zero init
__global__ void k_zero(float* __restrict__ out) {
  int i = blockIdx.x * blockDim.x + threadIdx.x;
  if (i < 4) out[OFF_ETOT + i] = 0.0f;
  if (i < N_ZERO_TAIL) out[OFF_FORCE + i] = 0.0f;
}

// ----------------------------------------------- pass 1: per-atom S0,S1 sums
// wave per atom; lanes stride over neighbors (coalesced 384B Rij chunks)
__global__ __launch_bounds__(256)
void k_pair_sum(const int* __restrict__ Nij, const int* __restrict__ Zij,
                const float* __restrict__ Rij, const int* __restrict__ emap,
                float* __restrict__ S /* [B*N][2] */) {
  int lane = threadIdx.x & 31;
  int aid  = blockIdx.x * 8 + (threadIdx.x >> 5);
  if (aid >= BATCH * N_ATOMS) return;
  const long base = (long)aid * MAX_NEI;
  float s0 = 0.0f, s1 = 0.0f;
  for (int m = lane; m < MAX_NEI; m += 32) {
    int nj = Nij[base + m];
    if (nj > 0) {
      const float* rp = Rij + (base + m) * 3;
      float x = rp[0], y = rp[1], z = rp[2];
      float r = sqrtf(x * x + y * y + z * z);
      if (r < CUTOFF) {
        float s = 0.5f * (cosf(PI_F * r / CUTOFF) + 1.0f);
        int  t = emap[Zij[base + m]];
        if (t == 0) s0 += s; else s1 += s;
      }
    }
  }
  #pragma unroll
  for (int o = 16; o >= 1; o >>= 1) {
    s0 += __shfl_xor(s0, o, 32);
    s1 += __shfl_xor(s1, o, 32);
  }
  if (lane == 0) { S[aid * 2 + 0] = s0; S[aid * 2 + 1] = s1; }
}

// ------------------------------------- MLP fwd + input-grad via f32 WMMA
// wave handles 16 atoms (M-dim of 16x16 tiles)
__global__ __launch_bounds__(256)
void k_mlp(const int* __restrict__ Zi, const int* __restrict__ emap,
           const float* __restrict__ S,
           const float* __restrict__ embC, const float* __restrict__ embN,
           const float* __restrict__ W1g, const float* __restrict__ b1g,
           const float* __restrict__ W2g, const float* __restrict__ b2g,
           float* __restrict__ out, float* __restrict__ Cpair /* [B*N][2] */) {
  __shared__ float sW1[D_EMB * D_HID];       // row-major (d, k): W1[d*64+k]
  __shared__ float sC[2 * D_EMB];            // emb_center
  __shared__ float sN[2 * D_EMB];            // emb_neigh
  __shared__ float sB1[D_HID];
  __shared__ float sW2[D_HID];
  __shared__ float sG[8][16 * D_HID];        // per-wave g = silu'(z)*W2 (16 atoms x 64)
  __shared__ int   sTC[8][16];               // per-wave center types

  const int tid  = threadIdx.x;
  const int w    = tid >> 5;
  const int lane = tid & 31;
  const int hl   = lane >> 4;                // half-wave (K/atom-group select)
  const int lm   = lane & 15;                // M index (atom) / N index (col)

  for (int i = tid; i < D_EMB * D_HID; i += 256) sW1[i] = W1g[i];
  if (tid < 64) { sC[tid] = embC[tid]; sN[tid] = embN[tid];
                  sB1[tid] = b1g[tid]; sW2[tid] = W2g[tid]; }

  const int waveG    = blockIdx.x * 8 + w;
  const int atombase = waveG * 16;
  if (lane < 16) sTC[w][lane] = emap[Zi[atombase + lane]];
  __syncthreads();

  // this lane's atom (per WMMA A layout: M = lane%16 for both halves)
  const float s0 = S[(atombase + lm) * 2 + 0];
  const float s1 = S[(atombase + lm) * 2 + 1];
  const float* Cc = &sC[sTC[w][lm] * D_EMB];

  // ---------- layer1 forward: Z(16x64) = H0(16x32) @ W1(32x64)
  v8f acc[4];
  #pragma unroll
  for (int nt = 0; nt < 4; ++nt) acc[nt] = v8f{};
  #pragma unroll
  for (int k = 0; k < 8; ++k) {
    const int kb = 4 * k + 2 * hl;
    v2f a;
    a.x = (s0 * sN[kb]     + s1 * sN[D_EMB + kb])     * Cc[kb];
    a.y = (s0 * sN[kb + 1] + s1 * sN[D_EMB + kb + 1]) * Cc[kb + 1];
    #pragma unroll
    for (int nt = 0; nt < 4; ++nt) {
      const int col = 16 * nt + lm;
      v2f bb;
      bb.x = sW1[kb * D_HID + col];
      bb.y = sW1[(kb + 1) * D_HID + col];
      acc[nt] = __builtin_amdgcn_wmma_f32_16x16x4_f32(
          false, a, false, bb, (short)0, acc[nt], false, false);
    }
  }

  // ---------- SiLU, Ei, and stage g = silu'(z)*W2 for backward
  // D layout: VGPR j of tile nt holds atom (j + 8*hl) at hid (16*nt + lm)
  const float b2v = b2g[0];
  float eip[8];
  #pragma unroll
  for (int j = 0; j < 8; ++j) eip[j] = 0.0f;
  #pragma unroll
  for (int nt = 0; nt < 4; ++nt) {
    const int col = 16 * nt + lm;
    const float b1v = sB1[col];
    const float w2v = sW2[col];
    #pragma unroll
    for (int j = 0; j < 8; ++j) {
      float z  = acc[nt][j] + b1v;
      float sg = 1.0f / (1.0f + expf(-z));
      eip[j] += (z * sg) * w2v;
      float dsilu = sg * (1.0f + z * (1.0f - sg));
      sG[w][(j + 8 * hl) * D_HID + col] = dsilu * w2v;
    }
  }
  #pragma unroll
  for (int o = 1; o < 16; o <<= 1)
    #pragma unroll
    for (int j = 0; j < 8; ++j) eip[j] += __shfl_xor(eip[j], o, 32);
  if (lm == 0) {
    float esum = 0.0f;
    #pragma unroll
    for (int j = 0; j < 8; ++j) {
      float e = eip[j] + b2v;
      out[OFF_EI + atombase + j + 8 * hl] = e;
      esum += e;
    }
    atomicAdd(&out[OFF_ETOT + atombase / N_ATOMS], esum);
  }
  __syncthreads();

  // ---------- backward: U(16x32) = Gmat(16x64) @ W1^T(64x32)
  v8f ua[2];
  ua[0] = v8f{}; ua[1] = v8f{};
  #pragma unroll
  for (int kc = 0; kc < 16; ++kc) {
    const int kb = 4 * kc + 2 * hl;
    v2f a;
    a.x = sG[w][lm * D_HID + kb];
    a.y = sG[w][lm * D_HID + kb + 1];
    #pragma unroll
    for (int nt = 0; nt < 2; ++nt) {
      const int d = 16 * nt + lm;
      v2f bb;                                  // W1T[kb][d] = W1[d][kb]
      bb.x = sW1[d * D_HID + kb];
      bb.y = sW1[d * D_HID + kb + 1];
      ua[nt] = __builtin_amdgcn_wmma_f32_16x16x4_f32(
          false, a, false, bb, (short)0, ua[nt], false, false);
    }
  }

  // c_t[m] = sum_d U[m,d] * embC[tc_m][d] * embN[t][d]
  float c0p[8], c1p[8];
  #pragma unroll
  for (int j = 0; j < 8; ++j) { c0p[j] = 0.0f; c1p[j] = 0.0f; }
  #pragma unroll
  for (int nt = 0; nt < 2; ++nt) {
    const int d = 16 * nt + lm;
    const float e0 = sN[d], e1 = sN[D_EMB + d];
    #pragma unroll
    for (int j = 0; j < 8; ++j) {
      const int m = j + 8 * hl;
      float q = ua[nt][j] * sC[sTC[w][m] * D_EMB + d];
      c0p[j] += q * e0;
      c1p[j] += q * e1;
    }
  }
  #pragma unroll
  for (int o = 1; o < 16; o <<= 1)
    #pragma unroll
    for (int j = 0; j < 8; ++j) {
      c0p[j] += __shfl_xor(c0p[j], o, 32);
      c1p[j] += __shfl_xor(c1p[j], o, 32);
    }
  if (lm == 0) {
    #pragma unroll
    for (int j = 0; j < 8; ++j) {
      const int m = j + 8 * hl;
      Cpair[(atombase + m) * 2 + 0] = c0p[j];
      Cpair[(atombase + m) * 2 + 1] = c1p[j];
    }
  }
}

// --------------------------- pass 2: per-pair forces + virial (wave per atom)
__global__ __launch_bounds__(256)
void k_force(const int* __restrict__ Nij, const int* __restrict__ Zij,
             const float* __restrict__ Rij, const int* __restrict__ emap,
             const float* __restrict__ Cpair, float* __restrict__ out) {
  __shared__ float sV[9];
  const int tid  = threadIdx.x;
  if (tid < 9) sV[tid] = 0.0f;
  __syncthreads();

  const int lane = tid & 31;
  const int aid  = blockIdx.x * 8 + (tid >> 5);   // global atom; 8 atoms/block, same batch
  const int b    = aid / N_ATOMS;
  const int ia   = aid - b * N_ATOMS;
  const float c0 = Cpair[aid * 2 + 0];
  const float c1 = Cpair[aid * 2 + 1];
  const long base = (long)aid * MAX_NEI;

  float fx = 0, fy = 0, fz = 0;
  float v[9];
  #pragma unroll
  for (int i = 0; i < 9; ++i) v[i] = 0.0f;

  for (int m = lane; m < MAX_NEI; m += 32) {
    int nj = Nij[base + m];
    if (nj <= 0) continue;
    const float* rp = Rij + (base + m) * 3;
    float x = rp[0], y = rp[1], z = rp[2];
    float r = sqrtf(x * x + y * y + z * z);
    if (!(r > 0.0f) || r >= CUTOFF) continue;      // masked / zero-gradient pairs
    int t = emap[Zij[base + m]];
    float dsdr = -0.5f * (PI_F / CUTOFF) * sinf(PI_F * r / CUTOFF);
    float dE   = (t == 0 ? c0 : c1) * dsdr;
    float coef = -dE / r;                          // dE_Rid = -dE * rij/r
    float gx = coef * x, gy = coef * y, gz = coef * z;
    fx += gx; fy += gy; fz += gz;
    float* Fj = out + OFF_FORCE + ((long)b * N_ALL + (nj - 1)) * 3;
    atomicAdd(Fj + 0, -gx);
    atomicAdd(Fj + 1, -gy);
    atomicAdd(Fj + 2, -gz);
    v[0] += x * gx; v[1] += x * gy; v[2] += x * gz;
    v[3] += y * gx; v[4] += y * gy; v[5] += y * gz;
    v[6] += z * gx; v[7] += z * gy; v[8] += z * gz;
  }

  #pragma unroll
  for (int o = 16; o >= 1; o >>= 1) {
    fx += __shfl_xor(fx, o, 32);
    fy += __shfl_xor(fy, o, 32);
    fz += __shfl_xor(fz, o, 32);
    #pragma unroll
    for (int i = 0; i < 9; ++i) v[i] += __shfl_xor(v[i], o, 32);
  }

  if (lane == 0) {
    float* Fi = out + OFF_FORCE + ((long)b * N_ALL + ia) * 3;
    atomicAdd(Fi + 0, fx);
    atomicAdd(Fi + 1, fy);
    atomicAdd(Fi + 2, fz);
    float* vp = out + OFF_VIRIAL_PER + (long)aid * 9;
    #pragma unroll
    for (int i = 0; i < 9; ++i) { vp[i] = v[i]; atomicAdd(&sV[i], v[i]); }
  }
  __syncthreads();
  if (tid < 9) atomicAdd(&out[OFF_VIRIAL + b * 9 + tid], sV[tid]);
}

// ---------------------------------------------------------------- launcher
extern "C" void kernel_launch(void* const* d_in, const int* in_sizes, int n_in,
                              void* d_out, int out_size, void* d_ws, size_t ws_size,
                              hipStream_t stream) {
  const int*   emap = (const int*)  d_in[0];
  const int*   Zi   = (const int*)  d_in[1];
  const int*   Nij  = (const int*)  d_in[2];
  const int*   Zij  = (const int*)  d_in[3];
  const float* Rij  = (const float*)d_in[4];
  // d_in[5] = n_ghost scalar (compile-time constant here)
  const float* embC = (const float*)d_in[6];
  const float* embN = (const float*)d_in[7];
  const float* W1   = (const float*)d_in[8];
  const float* b1   = (const float*)d_in[9];
  const float* W2   = (const float*)d_in[10];
  const float* b2   = (const float*)d_in[11];
  float* out = (float*)d_out;

  float* S     = (float*)d_ws;                 // [B*N][2]  pass-1 sums
  float* Cpair = S + BATCH * N_ATOMS * 2;      // [B*N][2]  backward coefficients

  k_zero<<<(N_ZERO_TAIL + 255) / 256, 256, 0, stream>>>(out);
  k_pair_sum<<<(BATCH * N_ATOMS) / 8, 256, 0, stream>>>(Nij, Zij, Rij, emap, S);
  k_mlp<<<(BATCH * N_ATOMS) / 16 / 8, 256, 0, stream>>>(Zi, emap, S, embC, embN,
                                                        W1, b1, W2, b2, out, Cpair);
  k_force<<<(BATCH * N_ATOMS) / 8, 256, 0, stream>>>(Nij, Zij, Rij, emap, Cpair, out);
}